// EquivarientScalar_44203803410681
// MI455X (gfx1250) — compile-verified
//
#include <hip/hip_runtime.h>
#include <math.h>

// ---------------------------------------------------------------------------
// Fused GatedEquivariant stack for MI455X (gfx1250), wave32 + WMMA f32.
//
// Roofline: ~65.5 GFLOP f32 vs ~256 MB HBM traffic -> memory bound (~11us at
// 23.3 TB/s). One fused pass over atoms; all GEMMs via V_WMMA_F32_16X16X4_F32;
// weights staged per block in LDS with GLOBAL_LOAD_ASYNC_TO_LDS_B128
// (ASYNCcnt) and L2-warmed with global_prefetch; K-loops software-pipelined
// so DS fragment loads run ahead of the dependent WMMAs.
// ---------------------------------------------------------------------------

#define N_ATOMS   100000
#define NF        128
#define NF2       256
#define NLAYERS   2
#define NGRAPHS   128

#define WAVES_PER_BLOCK 3
#define NTHREADS (WAVES_PER_BLOCK * 32)
#define TM 16
#define ATOMS_PER_BLOCK (WAVES_PER_BLOCK * TM)   // 48

#define WSTR  132    // padded LDS row stride for 128-wide tiles (bank spread)
#define SNSTR 260    // padded LDS row stride for 256-wide [s|n2] tile
#define WBUF_FLOATS (128 * WSTR)
#define PW_FLOATS   (48 * WSTR + 48 * WSTR + 16 * SNSTR + 16 * WSTR)
#define LDS_FLOATS  (WBUF_FLOATS + WAVES_PER_BLOCK * PW_FLOATS)
#define LDS_BYTES   (LDS_FLOATS * 4)             // 294912 B = 288 KB

typedef __attribute__((ext_vector_type(2))) float v2f;
typedef __attribute__((ext_vector_type(8))) float v8f;
typedef __attribute__((ext_vector_type(4))) int   v4i;

typedef __attribute__((address_space(1))) v4i gv4i;  // global
typedef __attribute__((address_space(3))) v4i lv4i;  // LDS

// ---------------- CDNA5 async copy: global -> LDS (ASYNCcnt) ----------------
__device__ __forceinline__ void async_ld_b128(float* lds_dst, const float* gsrc) {
#if defined(__has_builtin) && __has_builtin(__builtin_amdgcn_global_load_async_to_lds_b128)
  __builtin_amdgcn_global_load_async_to_lds_b128(
      (gv4i*)gsrc, (lv4i*)lds_dst, 0, 0);
#else
  asm volatile("global_load_async_to_lds_b128 %0, %1, off"
               :: "v"((unsigned)(unsigned long long)lds_dst), "v"(gsrc)
               : "memory");
#endif
}

__device__ __forceinline__ void wait_async0() {
#if defined(__has_builtin) && __has_builtin(__builtin_amdgcn_s_wait_asynccnt)
  __builtin_amdgcn_s_wait_asynccnt(0);
#else
  asm volatile("s_wait_asynccnt 0x0" ::: "memory");
#endif
}

// Stage a 128x128 f32 weight tile into LDS (stride WSTR) via async b128 copies.
__device__ __forceinline__ void stage_W(float* wbuf, const float* src, int ld_src,
                                        int tid) {
  for (int idx = tid; idx < 128 * 32; idx += NTHREADS) {
    const int r  = idx >> 5;
    const int c4 = (idx & 31) << 2;
    async_ld_b128(wbuf + r * WSTR + c4, src + (size_t)r * ld_src + c4);
  }
}

// Warm L2 with the next weight tile while WMMAs run (global_prefetch_b8).
__device__ __forceinline__ void prefetch_W(const float* src, int nfloats, int tid) {
  for (int off = tid * 32; off < nfloats; off += NTHREADS * 32)  // 128B lines
    __builtin_prefetch(src + off, 0, 0);
}

// ---------------- WMMA fragment loaders (LDS) -------------------------------
// A-fragment (16x4, MxK, f32): lane&15 -> row M, (lane>>4)*2 -> K offset
__device__ __forceinline__ v2f lds_ldA(const float* base, int stride, int k0, int lane) {
  const int row = lane & 15;
  const int k   = k0 + ((lane >> 4) << 1);
  const float* p = base + row * stride + k;
  v2f a; a.x = p[0]; a.y = p[1];
  return a;
}

// B-fragment (4x16, KxN, f32): lane&15 -> col N, (lane>>4)*2 -> K offset
__device__ __forceinline__ v2f lds_ldB(const float* wbuf, int k0, int n0, int lane) {
  const int col = n0 + (lane & 15);
  const int kb  = k0 + ((lane >> 4) << 1);
  v2f b; b.x = wbuf[kb * WSTR + col]; b.y = wbuf[(kb + 1) * WSTR + col];
  return b;
}

__device__ __forceinline__ v8f wmma4(v2f a, v2f b, v8f c) {
  // 8 args: (neg_a, A, neg_b, B, c_mod, C, reuse_a, reuse_b)
  return __builtin_amdgcn_wmma_f32_16x16x4_f32(false, a, false, b, (short)0, c,
                                               false, false);
}

// 3 spatial M-tiles x one 16-col tile over K=128, software-pipelined.
__device__ __forceinline__ void gemm3_col(const float* Vb, const float* wbuf,
                                          int n0, int lane,
                                          v8f& acc0, v8f& acc1, v8f& acc2) {
  v2f bf = lds_ldB(wbuf, 0, n0, lane);
  v2f a0 = lds_ldA(Vb + 0 * TM * WSTR, WSTR, 0, lane);
  v2f a1 = lds_ldA(Vb + 1 * TM * WSTR, WSTR, 0, lane);
  v2f a2 = lds_ldA(Vb + 2 * TM * WSTR, WSTR, 0, lane);
  for (int k0 = 0; k0 < NF; k0 += 4) {
    v2f bf_n = bf, a0n = a0, a1n = a1, a2n = a2;
    const int kn = k0 + 4;
    if (kn < NF) {                      // prefetch fragments for next step
      bf_n = lds_ldB(wbuf, kn, n0, lane);
      a0n = lds_ldA(Vb + 0 * TM * WSTR, WSTR, kn, lane);
      a1n = lds_ldA(Vb + 1 * TM * WSTR, WSTR, kn, lane);
      a2n = lds_ldA(Vb + 2 * TM * WSTR, WSTR, kn, lane);
    }
    acc0 = wmma4(a0, bf, acc0);
    acc1 = wmma4(a1, bf, acc1);
    acc2 = wmma4(a2, bf, acc2);
    bf = bf_n; a0 = a0n; a1 = a1n; a2 = a2n;
  }
}

// One M-tile x one 16-col tile over K=128, software-pipelined.
__device__ __forceinline__ void gemm1_col(const float* Ab, int astr,
                                          const float* wbuf, int n0, int lane,
                                          v8f& acc) {
  v2f bf = lds_ldB(wbuf, 0, n0, lane);
  v2f af = lds_ldA(Ab, astr, 0, lane);
  for (int k0 = 0; k0 < NF; k0 += 4) {
    v2f bf_n = bf, af_n = af;
    const int kn = k0 + 4;
    if (kn < NF) {
      bf_n = lds_ldB(wbuf, kn, n0, lane);
      af_n = lds_ldA(Ab, astr, kn, lane);
    }
    acc = wmma4(af, bf, acc);
    bf = bf_n; af = af_n;
  }
}

__global__ void __launch_bounds__(NTHREADS, 1)
gated_equiv_kernel(const float* __restrict__ s_in, const float* __restrict__ v_in,
                   const float* __restrict__ w1, const float* __restrict__ w2,
                   const float* __restrict__ a1w, const float* __restrict__ a1b,
                   const float* __restrict__ a2w, const float* __restrict__ a2b,
                   const float* __restrict__ out_w, const float* __restrict__ out_b,
                   float* __restrict__ sc_out)
{
  extern __shared__ float lds[];
  const int tid  = threadIdx.x;
  const int lane = tid & 31;
  const int wv   = tid >> 5;
  const int a0   = blockIdx.x * ATOMS_PER_BLOCK + wv * TM;

  float* wbuf = lds;                                   // 128 x WSTR weight tile
  float* Vb   = lds + WBUF_FLOATS + wv * PW_FLOATS;    // 48 x WSTR (3 spatial M-tiles)
  float* V1b  = Vb + 48 * WSTR;                        // 48 x WSTR
  float* SNb  = V1b + 48 * WSTR;                       // 16 x SNSTR ([s | n2])
  float* Hb   = SNb + 16 * SNSTR;                      // 16 x WSTR

  // Kick off staging of the first weight tile before streaming atom data.
  stage_W(wbuf, w2, NF, tid);

  // ------- stream in s (16x128) and v (3x16x128, spatial-major) -----------
  for (int r = 0; r < TM; ++r) {
    const int atom = a0 + r;
    float4 sv = make_float4(0.f, 0.f, 0.f, 0.f);
    if (atom < N_ATOMS) sv = *(const float4*)(s_in + (size_t)atom * NF + (lane << 2));
    *(float4*)(SNb + r * SNSTR + (lane << 2)) = sv;
  }
  for (int c = 0; c < 3; ++c) {
    for (int r = 0; r < TM; ++r) {
      const int atom = a0 + r;
      float4 vv = make_float4(0.f, 0.f, 0.f, 0.f);
      if (atom < N_ATOMS)
        vv = *(const float4*)(v_in + ((size_t)atom * 3 + c) * NF + (lane << 2));
      *(float4*)(Vb + (c * TM + r) * WSTR + (lane << 2)) = vv;
    }
  }

  for (int l = 0; l < NLAYERS; ++l) {
    const float* w2_l  = w2 + (size_t)l * NF * NF;
    const float* w1_l  = w1 + (size_t)l * NF * NF;
    const float* a1w_l = a1w + (size_t)l * NF2 * NF;
    const float* a2w_l = a2w + (size_t)l * NF * NF2;

    // ================= v2 = V @ w2 ; n2 = ||v2|| over spatial =============
    if (l > 0) {               // layer>0: stage w2_l here; prev iter only L2-prefetched it
      stage_W(wbuf, w2_l, NF, tid);
    }
    wait_async0();
    __syncthreads();
    prefetch_W(w1_l, NF * NF, tid);          // warm L2 for next stage
    for (int nt = 0; nt < 8; ++nt) {
      v8f acc0 = {}; v8f acc1 = {}; v8f acc2 = {};
      const int n0 = nt << 4;
      gemm3_col(Vb, wbuf, n0, lane, acc0, acc1, acc2);
      const int col   = n0 + (lane & 15);
      const int abase = (lane >> 4) << 3;
      #pragma unroll
      for (int i = 0; i < 8; ++i) {
        const float x = acc0[i], y = acc1[i], z = acc2[i];
        SNb[(abase + i) * SNSTR + NF + col] = sqrtf(x * x + y * y + z * z);
      }
    }
    // ================= v1 = V @ w1 (kept in LDS) ==========================
    __syncthreads();
    stage_W(wbuf, w1_l, NF, tid);
    wait_async0();
    __syncthreads();
    prefetch_W(a1w_l, NF2 * NF, tid);
    for (int nt = 0; nt < 8; ++nt) {
      v8f acc0 = {}; v8f acc1 = {}; v8f acc2 = {};
      const int n0 = nt << 4;
      gemm3_col(Vb, wbuf, n0, lane, acc0, acc1, acc2);
      const int col   = n0 + (lane & 15);
      const int abase = (lane >> 4) << 3;
      #pragma unroll
      for (int i = 0; i < 8; ++i) {
        V1b[(0 * TM + abase + i) * WSTR + col] = acc0[i];
        V1b[(1 * TM + abase + i) * WSTR + col] = acc1[i];
        V1b[(2 * TM + abase + i) * WSTR + col] = acc2[i];
      }
    }
    // ================= h = silu([s|n2] @ a1w + b1) ========================
    v8f hacc[8] = {};
    for (int p = 0; p < 2; ++p) {       // K=256 processed in two 128 halves
      __syncthreads();
      stage_W(wbuf, a1w_l + (size_t)p * NF * NF, NF, tid);
      wait_async0();
      __syncthreads();
      if (p == 0) prefetch_W(a1w_l + (size_t)NF * NF, NF * NF, tid);
      else        prefetch_W(a2w_l, NF * NF2, tid);
      for (int nt = 0; nt < 8; ++nt)
        gemm1_col(SNb + p * NF, SNSTR, wbuf, nt << 4, lane, hacc[nt]);
    }
    {
      const int abase = (lane >> 4) << 3;
      for (int nt = 0; nt < 8; ++nt) {
        const int col    = (nt << 4) + (lane & 15);
        const float bias = a1b[l * NF + col];
        #pragma unroll
        for (int i = 0; i < 8; ++i) {
          const float x = hacc[nt][i] + bias;
          Hb[(abase + i) * WSTR + col] = x / (1.f + __expf(-x));  // silu
        }
      }
    }
    // ====== [s_out | g] = h @ a2w + b2 ; s = s_out ; v = g * v1 ===========
    for (int q = 0; q < 2; ++q) {       // q=0: s_out cols, q=1: gate cols
      __syncthreads();
      stage_W(wbuf, a2w_l + q * NF, NF2, tid);
      wait_async0();
      __syncthreads();
      if (q == 1 && l + 1 < NLAYERS)
        prefetch_W(w2 + (size_t)(l + 1) * NF * NF, NF * NF, tid);
      for (int nt = 0; nt < 8; ++nt) {
        v8f acc = {};
        const int n0 = nt << 4;
        gemm1_col(Hb, WSTR, wbuf, n0, lane, acc);
        const int col    = n0 + (lane & 15);
        const float bias = a2b[l * NF2 + q * NF + col];
        const int abase  = (lane >> 4) << 3;
        #pragma unroll
        for (int i = 0; i < 8; ++i) {
          const float val = acc[i] + bias;
          const int a = abase + i;
          if (q == 0) {
            SNb[a * SNSTR + col] = val;                       // new s
          } else {
            Vb[(0 * TM + a) * WSTR + col] = val * V1b[(0 * TM + a) * WSTR + col];
            Vb[(1 * TM + a) * WSTR + col] = val * V1b[(1 * TM + a) * WSTR + col];
            Vb[(2 * TM + a) * WSTR + col] = val * V1b[(2 * TM + a) * WSTR + col];
          }
        }
      }
    }
  }

  // ================= sc = s @ out_w + out_b ===============================
  if (lane < TM) {
    const int atom = a0 + lane;
    if (atom < N_ATOMS) {
      float acc = out_b[0];
      for (int f = 0; f < NF; ++f) acc += SNb[lane * SNSTR + f] * out_w[f];
      sc_out[atom] = acc;
    }
  }
}

// y[b] = sum_n mask[b][n] * sc[n]   (one block per graph; mask streamed once,
// sc (400KB) stays hot in the 192MB L2 across blocks)
__global__ void __launch_bounds__(256, 1)
seg_sum_kernel(const float* __restrict__ mask, const float* __restrict__ sc,
               float* __restrict__ y)
{
  const int b = blockIdx.x;
  const float* mrow = mask + (size_t)b * N_ATOMS;
  float acc = 0.f;
  for (int n = threadIdx.x; n < N_ATOMS; n += blockDim.x)
    acc += mrow[n] * sc[n];
  for (int off = 16; off > 0; off >>= 1)
    acc += __shfl_down(acc, off, 32);
  __shared__ float red[8];
  if ((threadIdx.x & 31) == 0) red[threadIdx.x >> 5] = acc;
  __syncthreads();
  if (threadIdx.x < 8) {
    float r = red[threadIdx.x];
    for (int off = 4; off > 0; off >>= 1)
      r += __shfl_down(r, off, 32);
    if (threadIdx.x == 0) y[b] = r;
  }
}

extern "C" void kernel_launch(void* const* d_in, const int* in_sizes, int n_in,
                              void* d_out, int out_size, void* d_ws, size_t ws_size,
                              hipStream_t stream) {
  (void)in_sizes; (void)n_in; (void)out_size; (void)ws_size;
  const float* s    = (const float*)d_in[0];
  const float* v    = (const float*)d_in[1];
  /* d_in[2] = r, unused (dy=False path) */
  const float* mask = (const float*)d_in[3];
  const float* w1   = (const float*)d_in[4];
  const float* w2   = (const float*)d_in[5];
  const float* a1w  = (const float*)d_in[6];
  const float* a1b  = (const float*)d_in[7];
  const float* a2w  = (const float*)d_in[8];
  const float* a2b  = (const float*)d_in[9];
  const float* outw = (const float*)d_in[10];
  const float* outb = (const float*)d_in[11];
  float* y  = (float*)d_out;
  float* sc = (float*)d_ws;      // N_ATOMS floats of scratch

  // allow >64KB dynamic LDS (320KB/WGP on gfx1250); deterministic, no stream op
  (void)hipFuncSetAttribute((const void*)gated_equiv_kernel,
                            hipFuncAttributeMaxDynamicSharedMemorySize, LDS_BYTES);

  const int nblocks = (N_ATOMS + ATOMS_PER_BLOCK - 1) / ATOMS_PER_BLOCK;
  gated_equiv_kernel<<<nblocks, NTHREADS, LDS_BYTES, stream>>>(
      s, v, w1, w2, a1w, a1b, a2w, a2b, outw, outb, sc);
  seg_sum_kernel<<<NGRAPHS, 256, 0, stream>>>(mask, sc, y);
}